// GATNet_4810363372848
// MI455X (gfx1250) — compile-verified
//
#include <hip/hip_runtime.h>
#include <hip/hip_bf16.h>

typedef __attribute__((ext_vector_type(2))) float v2f;
typedef __attribute__((ext_vector_type(8))) float v8f;

#define NEG_SLOPE 0.2f

__device__ __forceinline__ void atomic_add_f32(float* p, float v) {
  // lowers to global/flat_atomic_add_f32 on gfx1250
  __hip_atomic_fetch_add(p, v, __ATOMIC_RELAXED, __HIP_MEMORY_SCOPE_AGENT);
}

// ---------------------------------------------------------------------------
// GEMM + attention logits: H[N,64] = X[N,K] @ W[K,64]
//   ssrc[n,h] = sum_c H[n, h*outc+c] * a_src[h*outc+c]   (heads*outc == 64)
//   sdst[n,h] = sum_c H[n, h*outc+c] * a_dst[h*outc+c]
// 256 threads = 8 waves; wave w computes rows [blk*128 + w*16, +16) x 64 cols
// via 4 accumulators of V_WMMA_F32_16X16X4_F32, K-loop step 4.
// ---------------------------------------------------------------------------
__global__ __launch_bounds__(256) void gat_gemm_logits(
    const float* __restrict__ X, const float* __restrict__ Wg,
    const float* __restrict__ a_src, const float* __restrict__ a_dst,
    float* __restrict__ H, float* __restrict__ ssrc, float* __restrict__ sdst,
    int N, int K, int heads, int outc)
{
  __shared__ float Wl[64][65];        // K-chunk of W, padded stride (bank-conflict free)
  __shared__ float hst[8][16][65];    // per-wave 16x64 output tile staging

  const int tid  = threadIdx.x;
  const int wid  = tid >> 5;
  const int lane = tid & 31;
  const int half = lane >> 4;         // 0: lanes 0-15, 1: lanes 16-31
  const int l16  = lane & 15;
  const int rowbase = blockIdx.x * 128 + wid * 16;

  // A-fragment row for this lane (16x4 f32 A: lanes hold M = lane%16); clamp for tail
  int ar = rowbase + l16;
  if (ar > N - 1) ar = N - 1;
  const size_t arowK = (size_t)ar * (size_t)K;

  v8f acc[4] = {};                    // 4 column tiles of 16

  for (int kc = 0; kc < K; kc += 64) {
    const int kcLen = (K - kc) < 64 ? (K - kc) : 64;   // both K=300 and K=64 are %4==0
    __syncthreads();
    for (int idx = tid; idx < kcLen * 64; idx += 256) {
      const int r = idx >> 6, c = idx & 63;
      Wl[r][c] = Wg[(size_t)(kc + r) * 64 + c];
    }
    __syncthreads();

    for (int kk = 0; kk < kcLen; kk += 4) {
      // A: VGPR0 = K{0,2}, VGPR1 = K{1,3} across lane halves -> contiguous pair per lane
      const int kb = kc + kk + 2 * half;
      v2f a;
      a.x = X[arowK + kb];
      a.y = X[arowK + kb + 1];

      const int rb = kk + 2 * half;   // B rows: same K pairing per lane half
      v2f b0, b1, b2, b3;
      b0.x = Wl[rb][l16];        b0.y = Wl[rb + 1][l16];
      b1.x = Wl[rb][16 + l16];   b1.y = Wl[rb + 1][16 + l16];
      b2.x = Wl[rb][32 + l16];   b2.y = Wl[rb + 1][32 + l16];
      b3.x = Wl[rb][48 + l16];   b3.y = Wl[rb + 1][48 + l16];

      acc[0] = __builtin_amdgcn_wmma_f32_16x16x4_f32(false, a, false, b0, (short)0, acc[0], false, false);
      acc[1] = __builtin_amdgcn_wmma_f32_16x16x4_f32(false, a, false, b1, (short)0, acc[1], false, false);
      acc[2] = __builtin_amdgcn_wmma_f32_16x16x4_f32(false, a, false, b2, (short)0, acc[2], false, false);
      acc[3] = __builtin_amdgcn_wmma_f32_16x16x4_f32(false, a, false, b3, (short)0, acc[3], false, false);
    }
  }

  // stage accumulators to LDS: C/D layout: VGPR v, lane L -> M = v + 8*(L/16), Nc = L%16
  #pragma unroll
  for (int ct = 0; ct < 4; ++ct)
    #pragma unroll
    for (int v = 0; v < 8; ++v)
      hst[wid][v + 8 * half][ct * 16 + l16] = acc[ct][v];
  __syncthreads();

  // coalesced H stores
  for (int r = 0; r < 16; ++r) {
    const int row = rowbase + r;
    if (row < N) {
      H[(size_t)row * 64 + lane]      = hst[wid][r][lane];
      H[(size_t)row * 64 + 32 + lane] = hst[wid][r][32 + lane];
    }
  }

  // per-node attention logits
  for (int idx = lane; idx < 16 * heads; idx += 32) {
    const int r = idx / heads, h = idx % heads;
    const int row = rowbase + r;
    if (row < N) {
      float s1 = 0.f, s2 = 0.f;
      for (int c = 0; c < outc; ++c) {
        const float hv = hst[wid][r][h * outc + c];
        s1 += hv * a_src[h * outc + c];
        s2 += hv * a_dst[h * outc + c];
      }
      ssrc[(size_t)row * heads + h] = s1;
      sdst[(size_t)row * heads + h] = s2;
    }
  }
}

// ---------------------------------------------------------------------------
// Per-(edge,head): numerator w = exp(leakyrelu(ssrc[src]+sdst[dst])),
// accumulate softmax denominator per (dst, head).
// ---------------------------------------------------------------------------
__global__ __launch_bounds__(256) void edge_softmax_num(
    const long long* __restrict__ src, const long long* __restrict__ dst,
    const float* __restrict__ ssrc, const float* __restrict__ sdst,
    float* __restrict__ wbuf, float* __restrict__ denom,
    long long total, int heads)
{
  const long long t = (long long)blockIdx.x * blockDim.x + threadIdx.x;
  if (t >= total) return;
  const long long e = t / heads;
  const int h = (int)(t - e * heads);
  const long long s = src[e], d = dst[e];
  float v = ssrc[s * heads + h] + sdst[d * heads + h];
  v = v > 0.f ? v : NEG_SLOPE * v;
  const float w = __expf(v);
  wbuf[t] = w;
  atomic_add_f32(&denom[d * heads + h], w);
}

// ---------------------------------------------------------------------------
// Per-(edge, channel i in 0..63): alpha = w / (denom + eps);
// Out[dst,i] += H[src,i] * alpha.  (head = i / outc)
// ---------------------------------------------------------------------------
__global__ __launch_bounds__(256) void edge_aggregate(
    const long long* __restrict__ src, const long long* __restrict__ dst,
    const float* __restrict__ H, const float* __restrict__ wbuf,
    const float* __restrict__ denom, float* __restrict__ Out,
    long long total, int heads, int outc)
{
  const long long t = (long long)blockIdx.x * blockDim.x + threadIdx.x;
  if (t >= total) return;
  const long long e = t >> 6;
  const int i = (int)(t & 63);
  const int h = i / outc;
  const long long s = src[e], d = dst[e];
  const float alpha = wbuf[e * heads + h] / (denom[d * heads + h] + 1e-16f);
  atomic_add_f32(&Out[d * 64 + i], H[s * 64 + i] * alpha);
}

__global__ __launch_bounds__(256) void fill_zero(float* __restrict__ p, long long n) {
  const long long t = (long long)blockIdx.x * blockDim.x + threadIdx.x;
  if (t < n) p[t] = 0.f;
}

// out[n*64+i] = b[i]  (pre-seed output with bias so atomics finish the job)
__global__ __launch_bounds__(256) void init_bias64(float* __restrict__ out,
                                                   const float* __restrict__ b, long long n) {
  const long long t = (long long)blockIdx.x * blockDim.x + threadIdx.x;
  if (t < n) out[t] = b[t & 63];
}

// in-place: a = elu(a + b[i%64])
__global__ __launch_bounds__(256) void bias_elu64(float* __restrict__ a,
                                                  const float* __restrict__ b, long long n) {
  const long long t = (long long)blockIdx.x * blockDim.x + threadIdx.x;
  if (t < n) {
    const float v = a[t] + b[t & 63];
    a[t] = v > 0.f ? v : (__expf(v) - 1.f);
  }
}

static inline unsigned blks(long long n, int b) { return (unsigned)((n + b - 1) / b); }

extern "C" void kernel_launch(void* const* d_in, const int* in_sizes, int n_in,
                              void* d_out, int out_size, void* d_ws, size_t ws_size,
                              hipStream_t stream) {
  const float*     x      = (const float*)d_in[0];
  const long long* ei     = (const long long*)d_in[1];   // int64 [2,E]
  const float*     W1     = (const float*)d_in[2];
  const float*     a_src1 = (const float*)d_in[3];
  const float*     a_dst1 = (const float*)d_in[4];
  const float*     b1     = (const float*)d_in[5];
  const float*     W2     = (const float*)d_in[6];
  const float*     a_src2 = (const float*)d_in[7];
  const float*     a_dst2 = (const float*)d_in[8];
  const float*     b2     = (const float*)d_in[9];
  float*           out    = (float*)d_out;

  const long long N = in_sizes[0] / 300;
  const long long E = in_sizes[1] / 2;
  const long long* src = ei;
  const long long* dst = ei + E;

  float* ws = (float*)d_ws;
  float* h1     = ws; ws += N * 64;
  float* ssrc1  = ws; ws += N * 8;
  float* sdst1  = ws; ws += N * 8;
  float* denom1 = ws; ws += N * 8;
  float* agg1   = ws; ws += N * 64;   // aggregated layer-1 output, then ELU'd in place
  float* h2     = ws; ws += N * 64;
  float* ssrc2  = ws; ws += N;
  float* sdst2  = ws; ws += N;
  float* denom2 = ws; ws += N;
  float* w1e    = ws; ws += E * 8;
  float* w2e    = ws; ws += E;

  // --- init accumulators ---
  fill_zero<<<blks(N * 8, 256), 256, 0, stream>>>(denom1, N * 8);
  fill_zero<<<blks(N * 64, 256), 256, 0, stream>>>(agg1, N * 64);
  fill_zero<<<blks(N, 256), 256, 0, stream>>>(denom2, N);
  init_bias64<<<blks(N * 64, 256), 256, 0, stream>>>(out, b2, N * 64);

  // --- layer 1: h1 = x @ W1, logits (heads=8, outc=8) ---
  gat_gemm_logits<<<blks(N, 128), 256, 0, stream>>>(
      x, W1, a_src1, a_dst1, h1, ssrc1, sdst1, (int)N, 300, 8, 8);
  edge_softmax_num<<<blks(E * 8, 256), 256, 0, stream>>>(
      src, dst, ssrc1, sdst1, w1e, denom1, E * 8, 8);
  edge_aggregate<<<blks(E * 64, 256), 256, 0, stream>>>(
      src, dst, h1, w1e, denom1, agg1, E * 64, 8, 8);
  bias_elu64<<<blks(N * 64, 256), 256, 0, stream>>>(agg1, b1, N * 64);

  // --- layer 2: h2 = act @ W2, logits (heads=1, outc=64) ---
  gat_gemm_logits<<<blks(N, 128), 256, 0, stream>>>(
      agg1, W2, a_src2, a_dst2, h2, ssrc2, sdst2, (int)N, 64, 1, 64);
  edge_softmax_num<<<blks(E, 256), 256, 0, stream>>>(
      src, dst, ssrc2, sdst2, w2e, denom2, E, 1);
  edge_aggregate<<<blks(E * 64, 256), 256, 0, stream>>>(
      src, dst, h2, w2e, denom2, out, E * 64, 1, 64);
}